// DualHeadGAT_71193377899244
// MI455X (gfx1250) — compile-verified
//
#include <hip/hip_runtime.h>

#define DEV_INLINE __device__ __forceinline__

typedef __attribute__((ext_vector_type(2))) float v2f;
typedef __attribute__((ext_vector_type(8))) float v8f;

// ---- order-preserving f32 <-> u32 mapping for atomic float max ----
DEV_INLINE unsigned f2ord(float f) {
    unsigned u = __float_as_uint(f);
    return (u & 0x80000000u) ? ~u : (u | 0x80000000u);
}
DEV_INLINE float ord2f(unsigned s) {
    return (s & 0x80000000u) ? __uint_as_float(s & 0x7fffffffu)
                             : __uint_as_float(~s);
}

// =====================================================================
// Generic fill (used to zero amax/denom/agg each call — harness poisons ws)
// =====================================================================
__global__ __launch_bounds__(256) void fill_u32(unsigned* __restrict__ p,
                                                unsigned v, long long n) {
    long long i = (long long)blockIdx.x * blockDim.x + threadIdx.x;
    if (i < n) p[i] = v;
}

// =====================================================================
// Layer 0 projection: h[n, c] = x[n,0]*W[0,c] + x[n,1]*W[1,c]   (F = 256)
// =====================================================================
__global__ __launch_bounds__(256) void gemm_k2(const float* __restrict__ x,
                                               const float* __restrict__ W,
                                               float* __restrict__ h, int N) {
    int idx = blockIdx.x * blockDim.x + threadIdx.x;
    if (idx >= N * 256) return;
    int n = idx >> 8;
    int c = idx & 255;
    h[idx] = fmaf(x[2 * n], W[c], x[2 * n + 1] * W[256 + c]);
}

// =====================================================================
// Layer 1 projection: C = A @ B with V_WMMA_F32_16X16X4_F32.
// One wave computes one 16x16 tile; K-loop steps 4.
// A-frag layout (ISA 7.12.2): lanes 0-15 -> M=lane, {K=k,k+1}; lanes 16-31 -> {K=k+2,k+3}
// B-frag layout: v0: lanes0-15 = row k cols0-15, lanes16-31 = row k+2; v1: rows k+1/k+3
// =====================================================================
__global__ __launch_bounds__(256) void gemm_wmma_f32(
    const float* __restrict__ A, const float* __restrict__ B,
    float* __restrict__ C, int M, int K, int Nc) {
    const int lane = threadIdx.x & 31;
    const int wave = threadIdx.x >> 5;
    const int tilesN = Nc >> 4;
    const int tile = blockIdx.x * (blockDim.x >> 5) + wave;
    const int tm = tile / tilesN;
    const int tn = tile - tm * tilesN;
    if (tm * 16 >= M) return;

    const int row = tm * 16 + (lane & 15);
    const int col = tn * 16 + (lane & 15);
    const int kh  = (lane >> 4) << 1;   // 0 (lanes 0-15) or 2 (lanes 16-31)

    const float* ap = A + (size_t)row * K + kh;    // 8B aligned (K=256, kh even)
    const float* bp = B + (size_t)kh * Nc + col;

    v8f acc = {};
    for (int k = 0; k < K; k += 4) {
        v2f a = *(const v2f*)(ap + k);             // A[row, k+kh .. k+kh+1]
        v2f b;
        b.x = bp[(size_t)k * Nc];                  // B[k+kh,   col]
        b.y = bp[(size_t)(k + 1) * Nc];            // B[k+kh+1, col]
        acc = __builtin_amdgcn_wmma_f32_16x16x4_f32(
            /*neg_a=*/false, a, /*neg_b=*/false, b,
            /*c_mod=*/(short)0, acc, /*reuse_a=*/false, /*reuse_b=*/false);
    }

    const int rbase = tm * 16 + ((lane >> 4) << 3);  // +0 or +8
    float* cp = C + (size_t)rbase * Nc + col;
#pragma unroll
    for (int i = 0; i < 8; ++i) cp[(size_t)i * Nc] = acc[i];
}

// =====================================================================
// Layer 2 projection: h2[n, 0..1] = feat[n, :] @ W2 (K=256, 2 outputs)
// wave per node, lane-parallel K, wave32 shuffle reduction
// =====================================================================
__global__ __launch_bounds__(256) void gemm_k256_n2(
    const float* __restrict__ in, const float* __restrict__ W2,
    float* __restrict__ h2, int N) {
    const int lane = threadIdx.x & 31;
    const int node = (blockIdx.x * blockDim.x + threadIdx.x) >> 5;
    if (node >= N) return;
    const float* ip = in + (size_t)node * 256;
    float s0 = 0.f, s1 = 0.f;
#pragma unroll
    for (int k = lane; k < 256; k += 32) {
        float v = ip[k];
        s0 = fmaf(v, W2[2 * k], s0);
        s1 = fmaf(v, W2[2 * k + 1], s1);
    }
#pragma unroll
    for (int off = 16; off > 0; off >>= 1) {
        s0 += __shfl_down(s0, off, 32);
        s1 += __shfl_down(s1, off, 32);
    }
    if (lane == 0) {
        h2[2 * node]     = s0;
        h2[2 * node + 1] = s1;
    }
}

// =====================================================================
// Per-node attention scores: esrc[n,h] = h[n,h,:]·a_src[h], same for dst
// =====================================================================
template <int H, int O>
__global__ __launch_bounds__(256) void node_scores(
    const float* __restrict__ h, const float* __restrict__ a_src,
    const float* __restrict__ a_dst, int N,
    float* __restrict__ esrc, float* __restrict__ edst) {
    const int idx = blockIdx.x * blockDim.x + threadIdx.x;
    if (idx >= N * H) return;
    const int n  = idx / H;
    const int hh = idx - n * H;
    const float* hp = h + (size_t)n * (H * O) + hh * O;
    float ss = 0.f, sd = 0.f;
#pragma unroll 8
    for (int o = 0; o < O; ++o) {
        float v = hp[o];
        ss = fmaf(v, a_src[hh * O + o], ss);
        sd = fmaf(v, a_dst[hh * O + o], sd);
    }
    esrc[idx] = ss;
    edst[idx] = sd;
}

// =====================================================================
// Edge pass A: alpha = leaky(esrc[s]+edst[d]); atomic segment-max over dst
// (edge ids >= E are the self-loops s=d=e-E)
// =====================================================================
template <int H>
__global__ __launch_bounds__(256) void edge_alpha_max(
    const int* __restrict__ ei, int E, int N,
    const float* __restrict__ esrc, const float* __restrict__ edst,
    float* __restrict__ alphaE, unsigned* __restrict__ amax) {
    const int e = blockIdx.x * blockDim.x + threadIdx.x;
    if (e >= E + N) return;
    int s, d;
    if (e < E) { s = ei[e]; d = ei[E + e]; } else { s = e - E; d = s; }
#pragma unroll
    for (int hh = 0; hh < H; ++hh) {
        float a = esrc[s * H + hh] + edst[d * H + hh];
        a = (a >= 0.f) ? a : 0.2f * a;
        alphaE[(size_t)e * H + hh] = a;
        atomicMax(&amax[d * H + hh], f2ord(a));
    }
}

// =====================================================================
// Edge pass B: ex = exp(alpha - amax[dst]); atomic segment-sum -> denom
// =====================================================================
template <int H>
__global__ __launch_bounds__(256) void edge_exp_sum(
    const int* __restrict__ ei, int E, int N,
    float* __restrict__ alphaE, const unsigned* __restrict__ amax,
    float* __restrict__ denom) {
    const int e = blockIdx.x * blockDim.x + threadIdx.x;
    if (e >= E + N) return;
    int d;
    if (e < E) { d = ei[E + e]; } else { d = e - E; }
#pragma unroll
    for (int hh = 0; hh < H; ++hh) {
        float m  = ord2f(amax[d * H + hh]);
        float ex = __expf(alphaE[(size_t)e * H + hh] - m);
        alphaE[(size_t)e * H + hh] = ex;
        atomicAdd(&denom[d * H + hh], ex);
    }
}

// =====================================================================
// Edge pass C: agg[dst] += h[src] * (ex / (denom[dst]+1e-16)); wave/edge,
// feature-parallel across lanes. Feature rows + accumulator are L2-resident.
// =====================================================================
template <int H, int O>
__global__ __launch_bounds__(256) void edge_aggregate(
    const int* __restrict__ ei, int E, int N,
    const float* __restrict__ h, const float* __restrict__ ex,
    const float* __restrict__ denom, float* __restrict__ agg) {
    const int lane = threadIdx.x & 31;
    const int e = (int)(((long long)blockIdx.x * blockDim.x + threadIdx.x) >> 5);
    if (e >= E + N) return;
    int s, d;
    if (e < E) { s = ei[e]; d = ei[E + e]; } else { s = e - E; d = s; }
    constexpr int F = H * O;
    float att[H];
#pragma unroll
    for (int hh = 0; hh < H; ++hh)
        att[hh] = ex[(size_t)e * H + hh] / (denom[d * H + hh] + 1e-16f);
    const float* hp = h + (size_t)s * F;
    float* ap = agg + (size_t)d * F;
    __builtin_prefetch(hp, 0, 3);   // global_prefetch_b8 on gfx1250
#pragma unroll
    for (int f = lane; f < F; f += 32)
        atomicAdd(&ap[f], hp[f] * att[f / O]);
}

// =====================================================================
// out = relu(agg + bias)   (F is 256 or 2 -> power of two)
// =====================================================================
template <int F>
__global__ __launch_bounds__(256) void bias_relu(
    const float* __restrict__ agg, const float* __restrict__ b,
    float* __restrict__ out, int N) {
    int idx = blockIdx.x * blockDim.x + threadIdx.x;
    if (idx >= N * F) return;
    out[idx] = fmaxf(agg[idx] + b[idx & (F - 1)], 0.f);
}

// =====================================================================
static inline dim3 gb(long long t) { return dim3((unsigned)((t + 255) / 256)); }

extern "C" void kernel_launch(void* const* d_in, const int* in_sizes, int n_in,
                              void* d_out, int out_size, void* d_ws, size_t ws_size,
                              hipStream_t stream) {
    (void)n_in; (void)out_size; (void)ws_size;
    const float* x   = (const float*)d_in[0];
    const int*   ei  = (const int*)d_in[1];
    const float* W0  = (const float*)d_in[2];
    const float* as0 = (const float*)d_in[3];
    const float* ad0 = (const float*)d_in[4];
    const float* b0  = (const float*)d_in[5];
    const float* W1  = (const float*)d_in[6];
    const float* as1 = (const float*)d_in[7];
    const float* ad1 = (const float*)d_in[8];
    const float* b1  = (const float*)d_in[9];
    const float* W2  = (const float*)d_in[10];
    const float* as2 = (const float*)d_in[11];
    const float* ad2 = (const float*)d_in[12];
    const float* b2  = (const float*)d_in[13];
    float* out = (float*)d_out;

    const int N    = in_sizes[0] / 2;   // x is [N, 2]
    const int E    = in_sizes[1] / 2;   // edge_index is [2, E]
    const int Etot = E + N;

    // ---- workspace carve ----
    float*    bufA   = (float*)d_ws;                         // [N,256] h0 / out0 / out1
    float*    bufB   = bufA   + (size_t)N * 256;             // [N,256] h1
    float*    bufAgg = bufB   + (size_t)N * 256;             // [N,256] accumulator
    float*    esrc   = bufAgg + (size_t)N * 256;             // [N,4]
    float*    edst   = esrc   + (size_t)N * 4;               // [N,4]
    unsigned* amax   = (unsigned*)(edst + (size_t)N * 4);    // [N,4]
    float*    denom  = (float*)(amax + (size_t)N * 4);       // [N,4]
    float*    alphaE = denom  + (size_t)N * 4;               // [Etot,4]
    float*    h2     = alphaE + (size_t)Etot * 4;            // [N,2]

    const int TPB = 256;
    const int wmma_waves  = ((N + 15) / 16) * (256 / 16);
    const int wmma_blocks = (wmma_waves + 7) / 8;

    // =============== Layer 0 (Fin=2, H=4, O=64, concat) ===============
    gemm_k2<<<gb((long long)N * 256), TPB, 0, stream>>>(x, W0, bufA, N);
    node_scores<4, 64><<<gb((long long)N * 4), TPB, 0, stream>>>(bufA, as0, ad0, N, esrc, edst);
    fill_u32<<<gb((long long)N * 4), TPB, 0, stream>>>(amax, 0u, (long long)N * 4);
    fill_u32<<<gb((long long)N * 4), TPB, 0, stream>>>((unsigned*)denom, 0u, (long long)N * 4);
    fill_u32<<<gb((long long)N * 256), TPB, 0, stream>>>((unsigned*)bufAgg, 0u, (long long)N * 256);
    edge_alpha_max<4><<<gb(Etot), TPB, 0, stream>>>(ei, E, N, esrc, edst, alphaE, amax);
    edge_exp_sum<4><<<gb(Etot), TPB, 0, stream>>>(ei, E, N, alphaE, amax, denom);
    edge_aggregate<4, 64><<<gb((long long)Etot * 32), TPB, 0, stream>>>(ei, E, N, bufA, alphaE, denom, bufAgg);
    bias_relu<256><<<gb((long long)N * 256), TPB, 0, stream>>>(bufAgg, b0, bufA, N);  // out0 -> bufA

    // =============== Layer 1 (Fin=256, H=4, O=64, concat) — WMMA GEMM ===============
    gemm_wmma_f32<<<dim3(wmma_blocks), TPB, 0, stream>>>(bufA, W1, bufB, N, 256, 256);
    node_scores<4, 64><<<gb((long long)N * 4), TPB, 0, stream>>>(bufB, as1, ad1, N, esrc, edst);
    fill_u32<<<gb((long long)N * 4), TPB, 0, stream>>>(amax, 0u, (long long)N * 4);
    fill_u32<<<gb((long long)N * 4), TPB, 0, stream>>>((unsigned*)denom, 0u, (long long)N * 4);
    fill_u32<<<gb((long long)N * 256), TPB, 0, stream>>>((unsigned*)bufAgg, 0u, (long long)N * 256);
    edge_alpha_max<4><<<gb(Etot), TPB, 0, stream>>>(ei, E, N, esrc, edst, alphaE, amax);
    edge_exp_sum<4><<<gb(Etot), TPB, 0, stream>>>(ei, E, N, alphaE, amax, denom);
    edge_aggregate<4, 64><<<gb((long long)Etot * 32), TPB, 0, stream>>>(ei, E, N, bufB, alphaE, denom, bufAgg);
    bias_relu<256><<<gb((long long)N * 256), TPB, 0, stream>>>(bufAgg, b1, bufA, N);  // out1 -> bufA

    // =============== Layer 2 (Fin=256, H=1, O=2, mean==identity) ===============
    gemm_k256_n2<<<gb((long long)N * 32), TPB, 0, stream>>>(bufA, W2, h2, N);
    node_scores<1, 2><<<gb((long long)N), TPB, 0, stream>>>(h2, as2, ad2, N, esrc, edst);
    fill_u32<<<gb((long long)N), TPB, 0, stream>>>(amax, 0u, (long long)N);
    fill_u32<<<gb((long long)N), TPB, 0, stream>>>((unsigned*)denom, 0u, (long long)N);
    fill_u32<<<gb((long long)N * 2), TPB, 0, stream>>>((unsigned*)bufAgg, 0u, (long long)N * 2);
    edge_alpha_max<1><<<gb(Etot), TPB, 0, stream>>>(ei, E, N, esrc, edst, alphaE, amax);
    edge_exp_sum<1><<<gb(Etot), TPB, 0, stream>>>(ei, E, N, alphaE, amax, denom);
    edge_aggregate<1, 2><<<gb((long long)Etot * 32), TPB, 0, stream>>>(ei, E, N, h2, alphaE, denom, bufAgg);
    bias_relu<2><<<gb((long long)N * 2), TPB, 0, stream>>>(bufAgg, b2, out, N);
}